// AMM_89945205113155
// MI455X (gfx1250) — compile-verified
//
#include <hip/hip_runtime.h>

typedef __attribute__((ext_vector_type(2))) float v2f;
typedef __attribute__((ext_vector_type(8))) float v8f;

#define BATCH  4
#define CHN    256
#define NPIX   4096
#define BN     (BATCH * NPIX)
#define NTILES (NPIX / 16)     // 256 j-tiles
#define SPLIT  8               // j-axis split factor (flash-decoding)
#define TPC    (NTILES / SPLIT) // tiles per chunk = 32

// ------------------------------------------------------------------
// Async global -> LDS copy (CDNA5), 16 bytes per lane, no VGPR data.
// ------------------------------------------------------------------
__device__ __forceinline__ void async_copy_b128(unsigned lds_byte_off,
                                                const float* gaddr) {
  asm volatile("global_load_async_to_lds_b128 %0, %1, off"
               :: "v"(lds_byte_off), "v"(gaddr)
               : "memory");
}

__device__ __forceinline__ void wait_async0() {
  asm volatile("s_wait_asynccnt 0x0" ::: "memory");
}

// ------------------------------------------------------------------
// Kernel 1: old_lambda / old_beta  (1x1 convs over channels)
// ------------------------------------------------------------------
__global__ __launch_bounds__(256) void k_lambda_beta(
    const float* __restrict__ ref, const float* __restrict__ wl,
    const float* __restrict__ bl,  const float* __restrict__ wb,
    const float* __restrict__ bb,  float* __restrict__ OL,
    float* __restrict__ OB)
{
  int idx = blockIdx.x * 256 + threadIdx.x;     // 0 .. B*N-1
  int b = idx >> 12;
  int n = idx & (NPIX - 1);
  const float* rp = ref + (size_t)b * CHN * NPIX + n;
  float sl = bl[0];
  float sb = bb[0];
#pragma unroll 8
  for (int c = 0; c < CHN; ++c) {
    float rv = rp[(size_t)c * NPIX];
    sl = fmaf(rv, wl[c], sl);
    sb = fmaf(rv, wb[c], sb);
  }
  OL[idx] = sl;
  OB[idx] = sb;
}

// ------------------------------------------------------------------
// Kernel 2: flash-attention over a 512-column j-chunk, f32 WMMA.
// grid = (N/64, B, SPLIT), block = 128 threads (4 waves).
// Wave w owns rows i0..i0+15; A tile resident in 128 VGPRs.
// B tile (16 j-cols x 256 ch) double-buffered through LDS via
// GLOBAL_LOAD_ASYNC_TO_LDS_B128. Writes per-row flash partials.
// ------------------------------------------------------------------
__global__ __launch_bounds__(128) void k_attn(
    const float* __restrict__ src, const float* __restrict__ ref,
    const float* __restrict__ OL,  const float* __restrict__ OB,
    float* __restrict__ PM, float* __restrict__ PL,
    float* __restrict__ PAL, float* __restrict__ PAB)
{
  __shared__ __align__(16) float lbuf[2][CHN * 16];   // 32 KB

  const int tid  = threadIdx.x;
  const int lane = tid & 31;
  const int wv   = tid >> 5;          // wave 0..3
  const int m    = lane & 15;         // A row / B col within fragment
  const int h    = lane >> 4;         // lane half -> K pair select
  const int b    = blockIdx.y;
  const int sp   = blockIdx.z;        // j-chunk 0..SPLIT-1
  const int i0   = blockIdx.x * 64 + wv * 16;
  const int jt0  = sp * TPC;

  const float* srcB = src + (size_t)b * CHN * NPIX;
  const float* refB = ref + (size_t)b * CHN * NPIX;

  // ---- load A tile (16 rows x 256 ch) into registers -------------
  // ISA A layout (16x4 f32): vgpr0 = K0(l0-15)/K2(l16-31), vgpr1 = K1/K3
  v2f a_frag[64];
  {
    const float* ap = srcB + i0 + m;
#pragma unroll
    for (int kk = 0; kk < 64; ++kk) {
      int c = 4 * kk + 2 * h;
      a_frag[kk].x = ap[(size_t)c * NPIX];
      a_frag[kk].y = ap[(size_t)(c + 1) * NPIX];
    }
  }

  // per-lane online-softmax state (one j-column slot per lane)
  float mrow[8], lrow[8], accL[8], accBt[8];
#pragma unroll
  for (int r = 0; r < 8; ++r) {
    mrow[r] = -3.0e38f; lrow[r] = 0.f; accL[r] = 0.f; accBt[r] = 0.f;
  }

  const int q  = tid & 3;    // which float4 in a 16-float channel row
  const int cb = tid >> 2;   // channel group 0..31

  // LDS byte offset of this thread's staging slot (buffer 0)
  const unsigned lds_slot0 =
      (unsigned)(uintptr_t)&lbuf[0][cb * 16 + 4 * q];
  const float* gstage = refB + (size_t)cb * NPIX + 4 * q;  // +32r*NPIX +16*jt

  // per-lane base for the B fragment reads
  const int frag_off = 2 * h * 16 + m;   // + 64*kk (+16 for odd channel)

  // ---- stage first tile of this chunk (async, register-free) -----
#pragma unroll
  for (int r = 0; r < 8; ++r) {
    async_copy_b128(lds_slot0 + (unsigned)(32 * r * 16 * 4),
                    gstage + (size_t)(32 * r) * NPIX + jt0 * 16);
  }
  wait_async0();
  __syncthreads();

#pragma unroll 1
  for (int jtl = 0; jtl < TPC; ++jtl) {
    const int jt  = jt0 + jtl;
    const int cur = jtl & 1;

    // kick off async staging of the next tile into the other buffer
    if (jtl + 1 < TPC) {
      const unsigned ldsn = lds_slot0 + (unsigned)((cur ^ 1) * CHN * 16 * 4);
      const float* gn = gstage + (jt + 1) * 16;
#pragma unroll
      for (int r = 0; r < 8; ++r) {
        async_copy_b128(ldsn + (unsigned)(32 * r * 16 * 4),
                        gn + (size_t)(32 * r) * NPIX);
      }
    }

    // ---- S = A x B : 64 chained V_WMMA_F32_16X16X4_F32 -----------
    const float* lb = &lbuf[cur][0] + frag_off;
    v8f acc = {0.f, 0.f, 0.f, 0.f, 0.f, 0.f, 0.f, 0.f};
    v2f bfA[4], bfB[4];
#pragma unroll
    for (int t = 0; t < 4; ++t) {
      bfA[t].x = lb[t * 64];
      bfA[t].y = lb[t * 64 + 16];
    }
#pragma unroll
    for (int g = 0; g < 16; ++g) {
      v2f* curf = (g & 1) ? bfB : bfA;
      v2f* nxtf = (g & 1) ? bfA : bfB;
      if (g < 15) {
#pragma unroll
        for (int t = 0; t < 4; ++t) {
          int kk = 4 * (g + 1) + t;
          nxtf[t].x = lb[kk * 64];
          nxtf[t].y = lb[kk * 64 + 16];
        }
      }
#pragma unroll
      for (int t = 0; t < 4; ++t) {
        acc = __builtin_amdgcn_wmma_f32_16x16x4_f32(
            false, a_frag[4 * g + t], false, curf[t], (short)0, acc,
            false, false);
      }
    }

    // ---- per-lane flash softmax update ---------------------------
    const int j0 = jt * 16;
    float lamj = OL[b * NPIX + j0 + m];
    float betj = OB[b * NPIX + j0 + m];
#pragma unroll
    for (int r = 0; r < 8; ++r) {
      float x  = acc[r];
      float mn = fmaxf(mrow[r], x);
      float sc = __expf(mrow[r] - mn);
      float p  = __expf(x - mn);
      lrow[r]  = lrow[r]  * sc + p;
      accL[r]  = accL[r]  * sc + p * lamj;
      accBt[r] = accBt[r] * sc + p * betj;
      mrow[r]  = mn;
    }

    wait_async0();
    __syncthreads();
  }

  // ---- flash-merge the 16 column lanes of each row; emit partials --
#pragma unroll
  for (int r = 0; r < 8; ++r) {
    float mm = mrow[r], ll = lrow[r], aL = accL[r], aB = accBt[r];
#pragma unroll
    for (int d = 1; d < 16; d <<= 1) {
      float m2 = __shfl_xor(mm, d, 32);
      float l2 = __shfl_xor(ll, d, 32);
      float a2 = __shfl_xor(aL, d, 32);
      float b2 = __shfl_xor(aB, d, 32);
      float mn = fmaxf(mm, m2);
      float s1 = __expf(mm - mn);
      float s2 = __expf(m2 - mn);
      ll = ll * s1 + l2 * s2;
      aL = aL * s1 + a2 * s2;
      aB = aB * s1 + b2 * s2;
      mm = mn;
    }
    if (m == 0) {
      int row = i0 + r + 8 * h;
      size_t o = ((size_t)(b * SPLIT + sp)) * NPIX + row;
      PM[o]  = mm;
      PL[o]  = ll;
      PAL[o] = aL;
      PAB[o] = aB;
    }
  }
}

// ------------------------------------------------------------------
// Kernel 2b: merge the SPLIT flash partials per row
// ------------------------------------------------------------------
__global__ __launch_bounds__(256) void k_merge(
    const float* __restrict__ PM,  const float* __restrict__ PL,
    const float* __restrict__ PAL, const float* __restrict__ PAB,
    float* __restrict__ NL, float* __restrict__ NB)
{
  int idx = blockIdx.x * 256 + threadIdx.x;   // 0 .. B*N-1
  int b = idx >> 12;
  int row = idx & (NPIX - 1);
  float mm = -3.0e38f, ll = 0.f, aL = 0.f, aB = 0.f;
#pragma unroll
  for (int s = 0; s < SPLIT; ++s) {
    size_t o = ((size_t)(b * SPLIT + s)) * NPIX + row;
    float m2 = PM[o], l2 = PL[o], a2 = PAL[o], b2 = PAB[o];
    float mn = fmaxf(mm, m2);
    float s1 = __expf(mm - mn);
    float s2 = __expf(m2 - mn);
    ll = ll * s1 + l2 * s2;
    aL = aL * s1 + a2 * s2;
    aB = aB * s1 + b2 * s2;
    mm = mn;
  }
  NL[idx] = aL / ll;
  NB[idx] = aB / ll;
}

// ------------------------------------------------------------------
// Kernel 3: out = new_lambda * src + new_beta  (broadcast over C)
// ------------------------------------------------------------------
__global__ __launch_bounds__(256) void k_mod(
    const float* __restrict__ src, const float* __restrict__ NL,
    const float* __restrict__ NB,  float* __restrict__ out)
{
  int idx = blockIdx.x * 256 + threadIdx.x;   // 0 .. B*C*N/4 - 1
  int n4 = idx & (NPIX / 4 - 1);
  int t  = idx >> 10;
  int c  = t & (CHN - 1);
  int b  = t >> 8;
  size_t off = ((size_t)(b * CHN + c)) * NPIX + (size_t)n4 * 4;
  float4 s  = *(const float4*)(src + off);
  float4 nl = *(const float4*)(NL + b * NPIX + n4 * 4);
  float4 nb = *(const float4*)(NB + b * NPIX + n4 * 4);
  float4 o;
  o.x = fmaf(nl.x, s.x, nb.x);
  o.y = fmaf(nl.y, s.y, nb.y);
  o.z = fmaf(nl.z, s.z, nb.z);
  o.w = fmaf(nl.w, s.w, nb.w);
  *(float4*)(out + off) = o;
}

// ------------------------------------------------------------------
extern "C" void kernel_launch(void* const* d_in, const int* in_sizes, int n_in,
                              void* d_out, int out_size, void* d_ws, size_t ws_size,
                              hipStream_t stream) {
  const float* src = (const float*)d_in[0];   // fm_source   [B,C,W,H]
  const float* ref = (const float*)d_in[1];   // fm_reference[B,C,W,H]
  const float* wl  = (const float*)d_in[2];   // w_lambda    [C]
  const float* bl  = (const float*)d_in[3];   // b_lambda    [1]
  const float* wb  = (const float*)d_in[4];   // w_beta      [C]
  const float* bb  = (const float*)d_in[5];   // b_beta      [1]
  float* out = (float*)d_out;

  float* ws = (float*)d_ws;
  float* OL  = ws;                       // [B,N]        old_lambda
  float* OB  = OL + BN;                  // [B,N]        old_beta
  float* NL  = OB + BN;                  // [B,N]        new_lambda
  float* NB  = NL + BN;                  // [B,N]        new_beta
  float* PM  = NB + BN;                  // [B,SPLIT,N]  partial max
  float* PL  = PM + BN * SPLIT;          // [B,SPLIT,N]  partial sum
  float* PAL = PL + BN * SPLIT;          // [B,SPLIT,N]  partial acc lambda
  float* PAB = PAL + BN * SPLIT;         // [B,SPLIT,N]  partial acc beta

  k_lambda_beta<<<BN / 256, 256, 0, stream>>>(ref, wl, bl, wb, bb, OL, OB);

  dim3 g2(NPIX / 64, BATCH, SPLIT);
  k_attn<<<g2, 128, 0, stream>>>(src, ref, OL, OB, PM, PL, PAL, PAB);

  k_merge<<<BN / 256, 256, 0, stream>>>(PM, PL, PAL, PAB, NL, NB);

  k_mod<<<(BATCH * CHN * NPIX / 4) / 256, 256, 0, stream>>>(src, NL, NB, out);
}